// DenseBlock_7035156431226
// MI455X (gfx1250) — compile-verified
//
#include <hip/hip_runtime.h>
#include <hip/hip_bf16.h>
#include <math.h>

// ---------------------------------------------------------------------------
// Sizes (compile-time constants from the reference)
// ---------------------------------------------------------------------------
#define BB 32
#define CCH 64
#define NNODE 512
#define TTIME 12
#define NLAYER 4
#define PAGES (BB * TTIME)                 // 384 (b,t) pages
#define PAGE_ELEMS (CCH * NNODE)           // 32768 floats per page
#define TENSOR ((size_t)PAGES * PAGE_ELEMS)  // 12,582,912 floats per tensor
#define BNT (BB * NNODE * TTIME)           // normalization count
#define EPSV 1e-5f

typedef float v2f __attribute__((ext_vector_type(2)));
typedef float v8f __attribute__((ext_vector_type(8)));

// fp32 WMMA: D(16x16) = A(16x4) * B(4x16) + C   (V_WMMA_F32_16X16X4_F32)
__device__ __forceinline__ v8f wmma4(v2f a, v2f b, v8f c) {
  return __builtin_amdgcn_wmma_f32_16x16x4_f32(false, a, false, b, (short)0, c,
                                               false, false);
}

__device__ __forceinline__ v8f vzero8() {
  v8f v;
#pragma unroll
  for (int i = 0; i < 8; ++i) v[i] = 0.f;
  return v;
}

__device__ __forceinline__ float sigm(float x) {
  return 1.f / (1.f + __expf(-x));
}

// ---------------------------------------------------------------------------
// Kernel 0: zero the BN-statistics accumulators (L * (sum, sumsq) * C)
// ---------------------------------------------------------------------------
__global__ void k_zero_stats(float* stats) {
  int i = threadIdx.x;
  if (i < NLAYER * 2 * CCH) stats[i] = 0.f;
}

// ---------------------------------------------------------------------------
// Kernel 1: transpose x[b,c,n,t] -> xt[b,t,c,n]  (page-major internal layout)
// ---------------------------------------------------------------------------
__global__ void k_transpose_in(const float* __restrict__ x,
                               float* __restrict__ xt) {
  size_t oi = (size_t)blockIdx.x * blockDim.x + threadIdx.x;
  if (oi >= TENSOR) return;
  int n = (int)(oi & (NNODE - 1));
  int c = (int)((oi >> 9) & (CCH - 1));
  int pg = (int)(oi >> 15);
  int t = pg % TTIME;
  int b = pg / TTIME;
  size_t ii = (((size_t)b * CCH + c) * NNODE + n) * TTIME + t;
  xt[oi] = x[ii];
}

// ---------------------------------------------------------------------------
// Kernel 2: per-channel sum / sumsq of h (btcn layout) -> stats[2][64]
// ---------------------------------------------------------------------------
__global__ void k_bn_stats(const float* __restrict__ h,
                           float* __restrict__ stats) {
  __shared__ float ssum[256], ssq[256];
  int tid = threadIdx.x;
  const float* p = h + (size_t)blockIdx.x * PAGE_ELEMS + (size_t)tid * 128;
  float s = 0.f, q = 0.f;
#pragma unroll 4
  for (int i = 0; i < 128; ++i) {
    float v = p[i];
    s += v;
    q += v * v;
  }
  ssum[tid] = s;
  ssq[tid] = q;
  __syncthreads();
  if (tid < CCH) {  // 4 consecutive threads per channel
    float ts = ssum[tid * 4] + ssum[tid * 4 + 1] + ssum[tid * 4 + 2] + ssum[tid * 4 + 3];
    float tq = ssq[tid * 4] + ssq[tid * 4 + 1] + ssq[tid * 4 + 2] + ssq[tid * 4 + 3];
    atomicAdd(&stats[tid], ts);
    atomicAdd(&stats[CCH + tid], tq);
  }
}

// ---------------------------------------------------------------------------
// Kernel 3: fused layer front-end, one block per (b,t) page, 128 thr = 4 waves
//   xb  = relu(BN(h))          (batch stats)
//   acc1 = RW@h + AW0@xb_prev + AW1@xb + (rb+ab)
//   acc2 = GW0@xb_prev + GW1@xb + gb
//   g    = tanh(acc1) * sigmoid(acc2)
//   u_e  = GC_e^T @ g   (e = 0..2);  hout = u0 + gcb,  u1/u2 -> workspace
// Two column tiles are processed per K-sweep so each weight A-fragment LDS
// load feeds two WMMAs.
// ---------------------------------------------------------------------------
#define WST 66  // padded LDS stride for 64x64 weight tiles

__global__ void __launch_bounds__(128) k_layer_front(
    const float* __restrict__ hin, float* __restrict__ hout,
    float* __restrict__ u1g, float* __restrict__ u2g,
    const float* __restrict__ stats, const float* __restrict__ gamma,
    const float* __restrict__ beta, const float* __restrict__ res_w,
    const float* __restrict__ res_b, const float* __restrict__ aff_w,
    const float* __restrict__ aff_b, const float* __restrict__ gate_w,
    const float* __restrict__ gate_b, const float* __restrict__ gc_w,
    const float* __restrict__ gc_b) {
  __shared__ float wres[CCH * WST], wa0[CCH * WST], wa1[CCH * WST];
  __shared__ float wg0[CCH * WST], wg1[CCH * WST];
  __shared__ float wc0[CCH * WST], wc1[CCH * WST], wc2[CCH * WST];
  __shared__ float bnsc[CCH], bnsh[CCH], bias1[CCH], bias2[CCH], biasc[CCH];
  __shared__ float gstage[4 * CCH * 16];

  int tid = threadIdx.x;
  int lane = tid & 31, wv = tid >> 5;
  int j = lane & 15, half = lane >> 4;
  int pg = blockIdx.x;
  int t = pg % TTIME;
  size_t base = (size_t)pg * PAGE_ELEMS;

  // stage weights (transposed where needed) into padded LDS
  for (int i = tid; i < CCH * CCH; i += 128) {
    int o = i >> 6, c = i & 63;
    int d = o * WST + c;
    wres[d] = res_w[o * CCH + c];
    wa0[d] = aff_w[(o * CCH + c) * 2 + 0];
    wa1[d] = aff_w[(o * CCH + c) * 2 + 1];
    wg0[d] = gate_w[(o * CCH + c) * 2 + 0];
    wg1[d] = gate_w[(o * CCH + c) * 2 + 1];
    wc0[d] = gc_w[(0 * CCH + c) * CCH + o];
    wc1[d] = gc_w[(1 * CCH + c) * CCH + o];
    wc2[d] = gc_w[(2 * CCH + c) * CCH + o];
  }
  if (tid < CCH) {
    const float invBNT = 1.f / (float)BNT;
    float mu = stats[tid] * invBNT;
    float var = stats[CCH + tid] * invBNT - mu * mu;
    float sc = gamma[tid] * rsqrtf(var + EPSV);
    bnsc[tid] = sc;
    bnsh[tid] = beta[tid] - mu * sc;
    bias1[tid] = res_b[tid] + aff_b[tid];
    bias2[tid] = gate_b[tid];
    biasc[tid] = gc_b[tid];
  }
  __syncthreads();

  const float* hp = hin + base;
  const float* hq = hin + base - PAGE_ELEMS;  // previous t page
  bool hasprev = (t != 0);
  float* gs = &gstage[wv * CCH * 16];

  // wave wv owns tiles [wv*8, wv*8+8); process them in pairs
  for (int q = 0; q < 4; ++q) {
    int n0a = (wv * 8 + q * 2) * 16;
    int n0b = n0a + 16;
    v8f acc1[2][4], acc2[2][4];
#pragma unroll
    for (int s = 0; s < 2; ++s)
#pragma unroll
      for (int mt = 0; mt < 4; ++mt) {
        acc1[s][mt] = vzero8();
        acc2[s][mt] = vzero8();
      }
    // K over 64 input channels, 4 at a time; A-fragments shared by both tiles
    for (int k = 0; k < 16; ++k) {
      int k0 = k * 4;
      int c0 = k0 + half, c1 = k0 + 2 + half;
      float sc0 = bnsc[c0], sh0 = bnsh[c0];
      float sc1 = bnsc[c1], sh1 = bnsh[c1];
      float ra0 = hp[c0 * NNODE + n0a + j];
      float ra1 = hp[c1 * NNODE + n0a + j];
      float rb0 = hp[c0 * NNODE + n0b + j];
      float rb1 = hp[c1 * NNODE + n0b + j];
      v2f braw[2], bxb[2], bxp[2];
      braw[0].x = ra0; braw[0].y = ra1;
      braw[1].x = rb0; braw[1].y = rb1;
      bxb[0].x = fmaxf(ra0 * sc0 + sh0, 0.f);
      bxb[0].y = fmaxf(ra1 * sc1 + sh1, 0.f);
      bxb[1].x = fmaxf(rb0 * sc0 + sh0, 0.f);
      bxb[1].y = fmaxf(rb1 * sc1 + sh1, 0.f);
      if (hasprev) {
        float pa0 = hq[c0 * NNODE + n0a + j];
        float pa1 = hq[c1 * NNODE + n0a + j];
        float pb0 = hq[c0 * NNODE + n0b + j];
        float pb1 = hq[c1 * NNODE + n0b + j];
        bxp[0].x = fmaxf(pa0 * sc0 + sh0, 0.f);
        bxp[0].y = fmaxf(pa1 * sc1 + sh1, 0.f);
        bxp[1].x = fmaxf(pb0 * sc0 + sh0, 0.f);
        bxp[1].y = fmaxf(pb1 * sc1 + sh1, 0.f);
      } else {
        bxp[0].x = 0.f; bxp[0].y = 0.f;
        bxp[1].x = 0.f; bxp[1].y = 0.f;
      }
      int ka = k0 + 2 * half;
#pragma unroll
      for (int mt = 0; mt < 4; ++mt) {
        int m = mt * 16 + (lane & 15);
        v2f a;
        a.x = wres[m * WST + ka];
        a.y = wres[m * WST + ka + 1];
        acc1[0][mt] = wmma4(a, braw[0], acc1[0][mt]);
        acc1[1][mt] = wmma4(a, braw[1], acc1[1][mt]);
        a.x = wa0[m * WST + ka];
        a.y = wa0[m * WST + ka + 1];
        acc1[0][mt] = wmma4(a, bxp[0], acc1[0][mt]);
        acc1[1][mt] = wmma4(a, bxp[1], acc1[1][mt]);
        a.x = wa1[m * WST + ka];
        a.y = wa1[m * WST + ka + 1];
        acc1[0][mt] = wmma4(a, bxb[0], acc1[0][mt]);
        acc1[1][mt] = wmma4(a, bxb[1], acc1[1][mt]);
        a.x = wg0[m * WST + ka];
        a.y = wg0[m * WST + ka + 1];
        acc2[0][mt] = wmma4(a, bxp[0], acc2[0][mt]);
        acc2[1][mt] = wmma4(a, bxp[1], acc2[1][mt]);
        a.x = wg1[m * WST + ka];
        a.y = wg1[m * WST + ka + 1];
        acc2[0][mt] = wmma4(a, bxb[0], acc2[0][mt]);
        acc2[1][mt] = wmma4(a, bxb[1], acc2[1][mt]);
      }
    }
    // per sub-tile: gate epilogue -> LDS staging, then u_e = GC_e^T @ g
#pragma unroll 1
    for (int s = 0; s < 2; ++s) {
      int n0 = s ? n0b : n0a;
#pragma unroll
      for (int mt = 0; mt < 4; ++mt) {
#pragma unroll
        for (int r = 0; r < 8; ++r) {
          int m = mt * 16 + r + 8 * half;
          float av = acc1[s][mt][r] + bias1[m];
          float gv = acc2[s][mt][r] + bias2[m];
          gs[m * 16 + j] = tanhf(av) * sigm(gv);
        }
      }
      v8f u0[4], u1a[4], u2a[4];
#pragma unroll
      for (int mt = 0; mt < 4; ++mt) {
        u0[mt] = vzero8();
        u1a[mt] = vzero8();
        u2a[mt] = vzero8();
      }
      for (int k = 0; k < 16; ++k) {
        int k0 = k * 4;
        v2f bg;
        bg.x = gs[(k0 + half) * 16 + j];
        bg.y = gs[(k0 + 2 + half) * 16 + j];
        int ka = k0 + 2 * half;
#pragma unroll
        for (int mt = 0; mt < 4; ++mt) {
          int m = mt * 16 + (lane & 15);
          v2f a;
          a.x = wc0[m * WST + ka];
          a.y = wc0[m * WST + ka + 1];
          u0[mt] = wmma4(a, bg, u0[mt]);
          a.x = wc1[m * WST + ka];
          a.y = wc1[m * WST + ka + 1];
          u1a[mt] = wmma4(a, bg, u1a[mt]);
          a.x = wc2[m * WST + ka];
          a.y = wc2[m * WST + ka + 1];
          u2a[mt] = wmma4(a, bg, u2a[mt]);
        }
      }
#pragma unroll
      for (int mt = 0; mt < 4; ++mt) {
#pragma unroll
        for (int r = 0; r < 8; ++r) {
          int m = mt * 16 + r + 8 * half;
          size_t idx = base + (size_t)m * NNODE + n0 + j;
          hout[idx] = u0[mt][r] + biasc[m];  // u0 + gc bias (graph kernel adds rest)
          u1g[idx] = u1a[mt][r];
          u2g[idx] = u2a[mt][r];
        }
      }
    }
  }
}

// ---------------------------------------------------------------------------
// Kernel 4: hout += u1 @ A0 + u2 @ A1, one block per (b,t) page, 8 waves.
// u_e page staged in LDS (padded, A-operand); supports streamed (L2-resident).
// K loop is OUTER, tile loop INNER: each LDS A-fragment feeds 4 WMMAs.
// ---------------------------------------------------------------------------
#define UST 516  // padded stride for staged 64x512 page

__global__ void __launch_bounds__(256) k_layer_graph(
    float* __restrict__ hio, const float* __restrict__ u1g,
    const float* __restrict__ u2g, const float* __restrict__ sup) {
  __shared__ float us[CCH * UST];  // 132 KB
  int tid = threadIdx.x;
  int lane = tid & 31, wv = tid >> 5;
  int j = lane & 15, half = lane >> 4;
  size_t base = (size_t)blockIdx.x * PAGE_ELEMS;

  v8f acc[4][4];
#pragma unroll
  for (int tt = 0; tt < 4; ++tt) {
    int n0 = (wv + tt * 8) * 16;
#pragma unroll
    for (int mt = 0; mt < 4; ++mt)
#pragma unroll
      for (int r = 0; r < 8; ++r)
        acc[tt][mt][r] = hio[base + (size_t)(mt * 16 + r + 8 * half) * NNODE + n0 + j];
  }

  for (int e = 0; e < 2; ++e) {
    const float* ug = e ? u2g : u1g;
    __syncthreads();
    for (int i = tid; i < PAGE_ELEMS; i += 256) {
      int c = i >> 9, n = i & (NNODE - 1);
      us[c * UST + n] = ug[base + i];
    }
    __syncthreads();
    const float* As = sup + (size_t)e * NNODE * NNODE;
    for (int k = 0; k < 128; ++k) {
      int k0 = k * 4;
      int ka = k0 + 2 * half;
      // A-fragments from LDS, shared across the 4 column tiles
      v2f afr[4];
#pragma unroll
      for (int mt = 0; mt < 4; ++mt) {
        int m = mt * 16 + (lane & 15);
        afr[mt].x = us[m * UST + ka];
        afr[mt].y = us[m * UST + ka + 1];
      }
      const float* Ar0 = As + (size_t)(k0 + half) * NNODE + j;
      const float* Ar1 = As + (size_t)(k0 + 2 + half) * NNODE + j;
#pragma unroll
      for (int tt = 0; tt < 4; ++tt) {
        int n0 = (wv + tt * 8) * 16;
        v2f b;
        b.x = Ar0[n0];
        b.y = Ar1[n0];
#pragma unroll
        for (int mt = 0; mt < 4; ++mt)
          acc[tt][mt] = wmma4(afr[mt], b, acc[tt][mt]);
      }
    }
  }
#pragma unroll
  for (int tt = 0; tt < 4; ++tt) {
    int n0 = (wv + tt * 8) * 16;
#pragma unroll
    for (int mt = 0; mt < 4; ++mt)
#pragma unroll
      for (int r = 0; r < 8; ++r)
        hio[base + (size_t)(mt * 16 + r + 8 * half) * NNODE + n0 + j] = acc[tt][mt][r];
  }
}

// ---------------------------------------------------------------------------
// Kernel 5: self-attention over the stack axis (5 tensors), WMMA MLP,
// softmax, weighted sum; writes final output in bcnt layout.
// ---------------------------------------------------------------------------
__global__ void __launch_bounds__(256) k_attention(
    const float* __restrict__ hstack, const float* __restrict__ p1w,
    const float* __restrict__ p1bv, const float* __restrict__ p2w,
    const float* __restrict__ p2bv, const float* __restrict__ p3w,
    const float* __restrict__ p3bv, float* __restrict__ out) {
  __shared__ float w1[CCH * WST], w2[CCH * WST];
  __shared__ float p3v[CCH], b1[CCH], b2[CCH];
  __shared__ float es[8 * CCH * 16], es2[8 * CCH * 16];

  int tid = threadIdx.x;
  int lane = tid & 31, wv = tid >> 5;
  int j = lane & 15, half = lane >> 4;
  int pg = blockIdx.x;
  int t = pg % TTIME, b = pg / TTIME;
  size_t base = (size_t)pg * PAGE_ELEMS;

  for (int i = tid; i < CCH * CCH; i += 256) {
    int o = i >> 6, c = i & 63;
    w1[o * WST + c] = p1w[i];
    w2[o * WST + c] = p2w[i];
  }
  if (tid < CCH) {
    p3v[tid] = p3w[tid];
    b1[tid] = p1bv[tid];
    b2[tid] = p2bv[tid];
  }
  __syncthreads();
  float p3bs = p3bv[0];
  float* esw = &es[wv * CCH * 16];
  float* es2w = &es2[wv * CCH * 16];

  for (int ct = wv; ct < 32; ct += 8) {
    int n0 = ct * 16;
    float e3v[5];
#pragma unroll 1
    for (int x = 0; x < 5; ++x) {
      const float* hx = hstack + (size_t)x * TENSOR + base;
      v8f a1[4];
#pragma unroll
      for (int mt = 0; mt < 4; ++mt) a1[mt] = vzero8();
      for (int k = 0; k < 16; ++k) {
        int k0 = k * 4;
        v2f bfrag;
        bfrag.x = fmaxf(hx[(k0 + half) * NNODE + n0 + j], 0.f);
        bfrag.y = fmaxf(hx[(k0 + 2 + half) * NNODE + n0 + j], 0.f);
        int ka = k0 + 2 * half;
#pragma unroll
        for (int mt = 0; mt < 4; ++mt) {
          int m = mt * 16 + (lane & 15);
          v2f a;
          a.x = w1[m * WST + ka];
          a.y = w1[m * WST + ka + 1];
          a1[mt] = wmma4(a, bfrag, a1[mt]);
        }
      }
#pragma unroll
      for (int mt = 0; mt < 4; ++mt)
#pragma unroll
        for (int r = 0; r < 8; ++r) {
          int m = mt * 16 + r + 8 * half;
          esw[m * 16 + j] = fmaxf(a1[mt][r] + b1[m], 0.f);
        }
      v8f a2[4];
#pragma unroll
      for (int mt = 0; mt < 4; ++mt) a2[mt] = vzero8();
      for (int k = 0; k < 16; ++k) {
        int k0 = k * 4;
        v2f bfrag;
        bfrag.x = esw[(k0 + half) * 16 + j];
        bfrag.y = esw[(k0 + 2 + half) * 16 + j];
        int ka = k0 + 2 * half;
#pragma unroll
        for (int mt = 0; mt < 4; ++mt) {
          int m = mt * 16 + (lane & 15);
          v2f a;
          a.x = w2[m * WST + ka];
          a.y = w2[m * WST + ka + 1];
          a2[mt] = wmma4(a, bfrag, a2[mt]);
        }
      }
#pragma unroll
      for (int mt = 0; mt < 4; ++mt)
#pragma unroll
        for (int r = 0; r < 8; ++r) {
          int m = mt * 16 + r + 8 * half;
          es2w[m * 16 + j] = fmaxf(a2[mt][r] + b2[m], 0.f);
        }
      float s = 0.f;
#pragma unroll
      for (int cc = 0; cc < 32; ++cc) {
        int c = half * 32 + cc;
        s += p3v[c] * es2w[c * 16 + j];
      }
      s += __shfl_xor(s, 16, 32);
      e3v[x] = s + p3bs;
    }
    // softmax over the 5 stack entries (per column j)
    float mx = e3v[0];
#pragma unroll
    for (int x = 1; x < 5; ++x) mx = fmaxf(mx, e3v[x]);
    float wx[5], wsum = 0.f;
#pragma unroll
    for (int x = 0; x < 5; ++x) {
      wx[x] = __expf(e3v[x] - mx);
      wsum += wx[x];
    }
    float inv = 1.f / wsum;
    // weighted sum and scatter back to bcnt layout
#pragma unroll 1
    for (int cc = 0; cc < 32; ++cc) {
      int c = cc * 2 + half;
      float accv = 0.f;
#pragma unroll
      for (int x = 0; x < 5; ++x) {
        const float* hx = hstack + (size_t)x * TENSOR + base;
        accv += wx[x] * fmaxf(hx[c * NNODE + n0 + j], 0.f);
      }
      out[(((size_t)b * CCH + c) * NNODE + (n0 + j)) * TTIME + t] = accv * inv;
    }
  }
}

// ---------------------------------------------------------------------------
// Host launcher
// ---------------------------------------------------------------------------
extern "C" void kernel_launch(void* const* d_in, const int* in_sizes, int n_in,
                              void* d_out, int out_size, void* d_ws,
                              size_t ws_size, hipStream_t stream) {
  const float* x = (const float*)d_in[0];
  const float* sup = (const float*)d_in[1];
  const float* bn_gamma = (const float*)d_in[2];
  const float* bn_beta = (const float*)d_in[3];
  const float* res_w = (const float*)d_in[4];
  const float* res_b = (const float*)d_in[5];
  const float* aff_w = (const float*)d_in[6];
  const float* aff_b = (const float*)d_in[7];
  const float* gate_w = (const float*)d_in[8];
  const float* gate_b = (const float*)d_in[9];
  const float* gc_w = (const float*)d_in[10];
  const float* gc_b = (const float*)d_in[11];
  const float* p1w = (const float*)d_in[12];
  const float* p1b = (const float*)d_in[13];
  const float* p2w = (const float*)d_in[14];
  const float* p2b = (const float*)d_in[15];
  const float* p3w = (const float*)d_in[16];
  const float* p3b = (const float*)d_in[17];

  float* W = (float*)d_ws;
  float* hstack = W;                       // 5 * TENSOR (x_t + 4 layer outs)
  float* u1 = W + 5 * TENSOR;              // TENSOR
  float* u2 = u1 + TENSOR;                 // TENSOR
  float* stats = u2 + TENSOR;              // L * 128

  k_zero_stats<<<1, 512, 0, stream>>>(stats);
  k_transpose_in<<<(unsigned)(TENSOR / 256), 256, 0, stream>>>(x, hstack);

  for (int l = 0; l < NLAYER; ++l) {
    float* hin = hstack + (size_t)l * TENSOR;
    float* hout = hstack + (size_t)(l + 1) * TENSOR;
    k_bn_stats<<<PAGES, 256, 0, stream>>>(hin, stats + l * 128);
    k_layer_front<<<PAGES, 128, 0, stream>>>(
        hin, hout, u1, u2, stats + l * 128, bn_gamma + l * CCH,
        bn_beta + l * CCH, res_w + l * CCH * CCH, res_b + l * CCH,
        aff_w + l * CCH * CCH * 2, aff_b + l * CCH, gate_w + l * CCH * CCH * 2,
        gate_b + l * CCH, gc_w + l * 3 * CCH * CCH, gc_b + l * CCH);
    k_layer_graph<<<PAGES, 256, 0, stream>>>(hout, u1, u2, sup);
  }

  k_attention<<<PAGES, 256, 0, stream>>>(hstack, p1w, p1b, p2w, p2b, p3w, p3b,
                                         (float*)d_out);
}